// GATBody_10024453669130
// MI455X (gfx1250) — compile-verified
//
#include <hip/hip_runtime.h>
#include <hip/hip_bf16.h>
#include <math.h>

#define NN    50000
#define EE    850000
#define FDIM  256      // = IN = H*D
#define NHEAD 4
#define HDIM  64
#define NEG_SLOPE 0.2f

typedef float v2f __attribute__((ext_vector_type(2)));
typedef float v8f __attribute__((ext_vector_type(8)));

// ---------------------------------------------------------------- CSR build
__global__ void k_zero(int* p, int n) {
  int i = blockIdx.x * 256 + threadIdx.x;
  if (i < n) p[i] = 0;
}

__global__ void k_count(const int* __restrict__ dst, int* __restrict__ cnt) {
  int e = blockIdx.x * 256 + threadIdx.x;
  if (e < EE) atomicAdd(&cnt[dst[e]], 1);
}

// single-block exclusive scan of cnt[0..NN) -> offsets; cursor (== cnt buffer)
// is reset to the exclusive prefix for the fill pass.
__global__ __launch_bounds__(1024) void k_scan(const int* cnt, int* offsets, int* cursor) {
  __shared__ int sm[1024];
  const int tid = threadIdx.x;
  int carry = 0;
  for (int base = 0; base < NN; base += 1024) {
    const int i = base + tid;
    const int v = (i < NN) ? cnt[i] : 0;
    sm[tid] = v;
    __syncthreads();
    for (int off = 1; off < 1024; off <<= 1) {
      int t = (tid >= off) ? sm[tid - off] : 0;
      __syncthreads();
      sm[tid] += t;
      __syncthreads();
    }
    const int excl = carry + sm[tid] - v;
    if (i < NN) { offsets[i] = excl; cursor[i] = excl; }
    carry += sm[1023];
    __syncthreads();
  }
  if (tid == 0) offsets[NN] = carry;
}

__global__ void k_fill(const int* __restrict__ dst, int* cursor, int* __restrict__ csr) {
  int e = blockIdx.x * 256 + threadIdx.x;
  if (e < EE) {
    int pdx = atomicAdd(&cursor[dst[e]], 1);
    csr[pdx] = e;
  }
}

// ---------------------------------------------------------------- weight pack
// Wpk[((k>>1)*256 + col)*2 + (k&1)] = W[k*256 + col]
// -> the two K-rows a lane needs for a B fragment become one aligned b64 load.
__global__ void k_pack(const float* __restrict__ W, float* __restrict__ Wpk) {
  int idx = blockIdx.x * 256 + threadIdx.x;   // k*256 + col
  if (idx >= FDIM * FDIM) return;
  const int k = idx >> 8, col = idx & 255;
  Wpk[(((k >> 1) << 8) + col) * 2 + (k & 1)] = W[idx];
}

// ---------------------------------------------------------------- WMMA GEMM
// C[N,256] = A[N,256] * W[256,256], f32 WMMA 16x16x4.
// block = 256 thr (8 waves) covering a 64-row x 256-col macro tile:
// wave w -> col tiles 2w, 2w+1; all 4 row tiles -> 8 accumulators / wave,
// 8 WMMA per K-step vs 4 A (b64) + 2 B (b64) loads.
__global__ __launch_bounds__(256) void k_gemm_wmma(const float* __restrict__ A,
                                                   const float* __restrict__ Wpk,
                                                   float* __restrict__ C) {
  const int lane  = threadIdx.x & 31;
  const int wave  = threadIdx.x >> 5;
  const int l15   = lane & 15;
  const int khalf = (lane >> 4) << 1;          // 0 or 2 (K split across lane halves)
  const int row0  = blockIdx.x << 6;           // 64 rows per block
  const int n0 = (wave << 5);                  // wave*2*16
  const int n1 = n0 + 16;

  const float* Ap0;
  const float* Ap1;
  const float* Ap2;
  const float* Ap3;
  {
    int r0 = row0 + l15;        r0 = (r0 < NN) ? r0 : (NN - 1);
    int r1 = row0 + 16 + l15;   r1 = (r1 < NN) ? r1 : (NN - 1);
    int r2 = row0 + 32 + l15;   r2 = (r2 < NN) ? r2 : (NN - 1);
    int r3 = row0 + 48 + l15;   r3 = (r3 < NN) ? r3 : (NN - 1);
    Ap0 = A + r0 * FDIM + khalf;
    Ap1 = A + r1 * FDIM + khalf;
    Ap2 = A + r2 * FDIM + khalf;
    Ap3 = A + r3 * FDIM + khalf;
  }
  // packed B pointers; per K-step of 4 the pair index advances by 2 -> +1024 floats
  const float* B0 = Wpk + ((khalf >> 1) << 9) + ((n0 + l15) << 1);
  const float* B1 = B0 + 32;

  v8f c0 = {}, c1 = {}, c2 = {}, c3 = {}, c4 = {}, c5 = {}, c6 = {}, c7 = {};
  for (int k0 = 0; k0 < FDIM; k0 += 4) {
    v2f b0 = *(const v2f*)B0;  B0 += 1024;
    v2f b1 = *(const v2f*)B1;  B1 += 1024;
    v2f a0 = *(const v2f*)(Ap0 + k0);
    v2f a1 = *(const v2f*)(Ap1 + k0);
    v2f a2 = *(const v2f*)(Ap2 + k0);
    v2f a3 = *(const v2f*)(Ap3 + k0);
    c0 = __builtin_amdgcn_wmma_f32_16x16x4_f32(false, a0, false, b0, (short)0, c0, false, false);
    c1 = __builtin_amdgcn_wmma_f32_16x16x4_f32(false, a0, false, b1, (short)0, c1, false, false);
    c2 = __builtin_amdgcn_wmma_f32_16x16x4_f32(false, a1, false, b0, (short)0, c2, false, false);
    c3 = __builtin_amdgcn_wmma_f32_16x16x4_f32(false, a1, false, b1, (short)0, c3, false, false);
    c4 = __builtin_amdgcn_wmma_f32_16x16x4_f32(false, a2, false, b0, (short)0, c4, false, false);
    c5 = __builtin_amdgcn_wmma_f32_16x16x4_f32(false, a2, false, b1, (short)0, c5, false, false);
    c6 = __builtin_amdgcn_wmma_f32_16x16x4_f32(false, a3, false, b0, (short)0, c6, false, false);
    c7 = __builtin_amdgcn_wmma_f32_16x16x4_f32(false, a3, false, b1, (short)0, c7, false, false);
  }

  const int rhalf = (lane >> 4) << 3;          // VGPR r holds rows m0+r / m0+8+r
  v8f* accs[8] = {&c0, &c1, &c2, &c3, &c4, &c5, &c6, &c7};
#pragma unroll
  for (int t = 0; t < 4; ++t) {
    const int rbase = row0 + t * 16 + rhalf;
    v8f& ca = *accs[2 * t];
    v8f& cb = *accs[2 * t + 1];
#pragma unroll
    for (int r = 0; r < 8; ++r) {
      const int row = rbase + r;
      if (row < NN) {
        C[row * FDIM + n0 + l15] = ca[r];
        C[row * FDIM + n1 + l15] = cb[r];
      }
    }
  }
}

// ---------------------------------------------------------------- attention logits
// el[n,h] = sum_d ft[n,h,d]*al[h,d];  er likewise. one thread per (n,h).
__global__ void k_elr(const float* __restrict__ ft, const float* __restrict__ al,
                      const float* __restrict__ ar, float* __restrict__ el,
                      float* __restrict__ er) {
  int idx = blockIdx.x * 256 + threadIdx.x;
  if (idx >= NN * NHEAD) return;
  const int n = idx >> 2, h = idx & 3;
  const float* f = ft + n * FDIM + h * HDIM;
  const float* a = al + h * HDIM;
  const float* r = ar + h * HDIM;
  float sl = 0.f, sr = 0.f;
#pragma unroll
  for (int d = 0; d < HDIM; d += 4) {
    float4 fv = *(const float4*)(f + d);
    float4 av = *(const float4*)(a + d);
    float4 rv = *(const float4*)(r + d);
    sl += fv.x * av.x + fv.y * av.y + fv.z * av.z + fv.w * av.w;
    sr += fv.x * rv.x + fv.y * rv.y + fv.z * rv.z + fv.w * rv.w;
  }
  el[idx] = sl;
  er[idx] = sr;
}

// ---------------------------------------------------------------- fused edge-softmax + aggregate
// one wave32 per destination node; lane covers d = 2*lane, 2*lane+1 for all 4 heads.
template <bool RES, bool ACT, bool MEAN>
__global__ __launch_bounds__(256) void k_agg(const float* __restrict__ ft,
                                             const float* __restrict__ el,
                                             const float* __restrict__ er,
                                             const int* __restrict__ offsets,
                                             const int* __restrict__ csr,
                                             const int* __restrict__ srcIdx,
                                             const float* __restrict__ hin,
                                             const float* __restrict__ bias,
                                             float* __restrict__ out) {
  const int node = blockIdx.x * 8 + (threadIdx.x >> 5);
  const int lane = threadIdx.x & 31;
  if (node >= NN) return;
  const float4 erv = *(const float4*)(er + node * 4);
  const int beg = offsets[node];
  const int end = offsets[node + 1];

  // phase 1: edge-parallel max of leaky_relu(el[src]+er[dst]) per head
  float m0 = -1e30f, m1 = -1e30f, m2 = -1e30f, m3 = -1e30f;
  for (int i = beg + lane; i < end; i += 32) {
    const int e = csr[i];
    const int s = srcIdx[e];
    const float4 q = *(const float4*)(el + s * 4);
    float v;
    v = q.x + erv.x; v = (v > 0.f) ? v : NEG_SLOPE * v; m0 = fmaxf(m0, v);
    v = q.y + erv.y; v = (v > 0.f) ? v : NEG_SLOPE * v; m1 = fmaxf(m1, v);
    v = q.z + erv.z; v = (v > 0.f) ? v : NEG_SLOPE * v; m2 = fmaxf(m2, v);
    v = q.w + erv.w; v = (v > 0.f) ? v : NEG_SLOPE * v; m3 = fmaxf(m3, v);
  }
#pragma unroll
  for (int off = 16; off > 0; off >>= 1) {
    m0 = fmaxf(m0, __shfl_xor(m0, off, 32));
    m1 = fmaxf(m1, __shfl_xor(m1, off, 32));
    m2 = fmaxf(m2, __shfl_xor(m2, off, 32));
    m3 = fmaxf(m3, __shfl_xor(m3, off, 32));
  }

  // phase 2: uniform edge loop; accumulate ft[src]*exp(e-m) and denominator
  float a0x = 0.f, a0y = 0.f, a1x = 0.f, a1y = 0.f;
  float a2x = 0.f, a2y = 0.f, a3x = 0.f, a3y = 0.f;
  float d0 = 0.f, d1 = 0.f, d2 = 0.f, d3 = 0.f;
  const int db = lane << 1;
  for (int i = beg; i < end; ++i) {
    const int e = csr[i];
    const int s = srcIdx[e];
    const float4 q = *(const float4*)(el + s * 4);
    const float* fp = ft + s * FDIM + db;
    float v, ex;
    float2 f;
    v = q.x + erv.x; v = (v > 0.f) ? v : NEG_SLOPE * v; ex = __expf(v - m0); d0 += ex;
    f = *(const float2*)(fp);            a0x += f.x * ex; a0y += f.y * ex;
    v = q.y + erv.y; v = (v > 0.f) ? v : NEG_SLOPE * v; ex = __expf(v - m1); d1 += ex;
    f = *(const float2*)(fp + HDIM);     a1x += f.x * ex; a1y += f.y * ex;
    v = q.z + erv.z; v = (v > 0.f) ? v : NEG_SLOPE * v; ex = __expf(v - m2); d2 += ex;
    f = *(const float2*)(fp + 2 * HDIM); a2x += f.x * ex; a2y += f.y * ex;
    v = q.w + erv.w; v = (v > 0.f) ? v : NEG_SLOPE * v; ex = __expf(v - m3); d3 += ex;
    f = *(const float2*)(fp + 3 * HDIM); a3x += f.x * ex; a3y += f.y * ex;
  }

  auto finish = [&](float ax, float ay, float dh, int h, float& vx, float& vy) {
    const float inv = 1.0f / fmaxf(dh, 1e-9f);
    vx = ax * inv;
    vy = ay * inv;
    if (RES) {
      const float2 rv = *(const float2*)(hin + node * FDIM + h * HDIM + db);
      vx += rv.x; vy += rv.y;
    }
    const float2 bv = *(const float2*)(bias + h * HDIM + db);
    vx += bv.x; vy += bv.y;
    if (ACT) {
      vx = (vx > 0.f) ? vx : expm1f(vx);
      vy = (vy > 0.f) ? vy : expm1f(vy);
    }
  };

  float vx0, vy0, vx1, vy1, vx2, vy2, vx3, vy3;
  finish(a0x, a0y, d0, 0, vx0, vy0);
  finish(a1x, a1y, d1, 1, vx1, vy1);
  finish(a2x, a2y, d2, 2, vx2, vy2);
  finish(a3x, a3y, d3, 3, vx3, vy3);

  if (MEAN) {
    *(float2*)(out + node * HDIM + db) =
        make_float2((vx0 + vx1 + vx2 + vx3) * 0.25f, (vy0 + vy1 + vy2 + vy3) * 0.25f);
  } else {
    float* o = out + node * FDIM + db;
    *(float2*)(o)            = make_float2(vx0, vy0);
    *(float2*)(o + HDIM)     = make_float2(vx1, vy1);
    *(float2*)(o + 2 * HDIM) = make_float2(vx2, vy2);
    *(float2*)(o + 3 * HDIM) = make_float2(vx3, vy3);
  }
}

// ---------------------------------------------------------------- driver
extern "C" void kernel_launch(void* const* d_in, const int* in_sizes, int n_in,
                              void* d_out, int out_size, void* d_ws, size_t ws_size,
                              hipStream_t stream) {
  (void)in_sizes; (void)n_in; (void)out_size; (void)ws_size;
  const float* x   = (const float*)d_in[0];
  const int*   src = (const int*)d_in[1];
  const int*   dst = (const int*)d_in[2];
  const float *W[3], *al[3], *ar[3], *b[3];
  for (int i = 0; i < 3; ++i) {
    W[i]  = (const float*)d_in[3 + 4 * i];
    al[i] = (const float*)d_in[4 + 4 * i];
    ar[i] = (const float*)d_in[5 + 4 * i];
    b[i]  = (const float*)d_in[6 + 4 * i];
  }

  char* p = (char*)d_ws;
  auto carve = [&](size_t bytes) -> char* {
    char* q = p;
    p += (bytes + 255) & ~(size_t)255;
    return q;
  };
  int*   offsets = (int*)carve((NN + 1) * sizeof(int));
  int*   cnt     = (int*)carve(NN * sizeof(int));        // reused as fill cursor
  int*   csr     = (int*)carve((size_t)EE * sizeof(int));
  float* el      = (float*)carve((size_t)NN * NHEAD * sizeof(float));
  float* er      = (float*)carve((size_t)NN * NHEAD * sizeof(float));
  float* Wpk     = (float*)carve((size_t)FDIM * FDIM * sizeof(float));
  float* ft      = (float*)carve((size_t)NN * FDIM * sizeof(float));
  float* h1      = (float*)carve((size_t)NN * FDIM * sizeof(float));
  float* h2      = (float*)carve((size_t)NN * FDIM * sizeof(float));

  // build CSR (dst -> incoming edge list); graph is identical every launch
  k_zero<<<(NN + 255) / 256, 256, 0, stream>>>(cnt, NN);
  k_count<<<(EE + 255) / 256, 256, 0, stream>>>(dst, cnt);
  k_scan<<<1, 1024, 0, stream>>>(cnt, offsets, cnt);
  k_fill<<<(EE + 255) / 256, 256, 0, stream>>>(dst, cnt, csr);

  const int gemmBlocks = (NN + 63) / 64;            // 782 (64-row macro tiles)
  const int packBlocks = (FDIM * FDIM) / 256;       // 256
  const int aggBlocks  = NN / 8;                    // 6250
  const int elrBlocks  = (NN * NHEAD + 255) / 256;

  // layer 0: no residual, ELU
  k_pack<<<packBlocks, 256, 0, stream>>>(W[0], Wpk);
  k_gemm_wmma<<<gemmBlocks, 256, 0, stream>>>(x, Wpk, ft);
  k_elr<<<elrBlocks, 256, 0, stream>>>(ft, al[0], ar[0], el, er);
  k_agg<false, true, false><<<aggBlocks, 256, 0, stream>>>(ft, el, er, offsets, csr, src,
                                                           nullptr, b[0], h1);
  // layer 1: residual, ELU
  k_pack<<<packBlocks, 256, 0, stream>>>(W[1], Wpk);
  k_gemm_wmma<<<gemmBlocks, 256, 0, stream>>>(h1, Wpk, ft);
  k_elr<<<elrBlocks, 256, 0, stream>>>(ft, al[1], ar[1], el, er);
  k_agg<true, true, false><<<aggBlocks, 256, 0, stream>>>(ft, el, er, offsets, csr, src,
                                                          h1, b[1], h2);
  // layer 2: residual, no act, fused head-mean -> d_out [N, 64]
  k_pack<<<packBlocks, 256, 0, stream>>>(W[2], Wpk);
  k_gemm_wmma<<<gemmBlocks, 256, 0, stream>>>(h2, Wpk, ft);
  k_elr<<<elrBlocks, 256, 0, stream>>>(ft, al[2], ar[2], el, er);
  k_agg<true, false, true><<<aggBlocks, 256, 0, stream>>>(ft, el, er, offsets, csr, src,
                                                          h2, b[2], (float*)d_out);
}